// LESL_71021579206675
// MI455X (gfx1250) — compile-verified
//
#include <hip/hip_runtime.h>
#include <hip/hip_bf16.h>

typedef __attribute__((ext_vector_type(2))) float v2f;
typedef __attribute__((ext_vector_type(8))) float v8f;

#define D_DIM    128
#define N_LANGS  8000
#define BLK_I    64
#define BLK_J    32
#define ROWPAD   132   // 128 + 4 floats: shifts rows by 4 banks -> conflict-free

// Async copy of one 16B chunk from global (base + voff) into LDS at lds_dst.
// Portable inline-asm path (CDNA5 GLOBAL_LOAD_ASYNC_TO_LDS_B128, GVS mode:
// saddr = uniform base, vaddr = per-lane 32-bit byte offset). Tracked by ASYNCcnt.
__device__ __forceinline__ void async_copy16(const float* lds_dst, unsigned voff,
                                             const float* base) {
    unsigned lds_off = (unsigned)(unsigned long long)lds_dst; // low 32b = LDS offset
    asm volatile("global_load_async_to_lds_b128 %0, %1, %2"
                 :
                 : "v"(lds_off), "v"(voff), "s"(base)
                 : "memory");
}

__device__ __forceinline__ void wait_async0() {
#if __has_builtin(__builtin_amdgcn_s_wait_asynccnt)
    __builtin_amdgcn_s_wait_asynccnt(0);
#else
    asm volatile("s_wait_asynccnt 0x0" ::: "memory");
#endif
}

// Kernel 1: each block computes partial (sum_contrib, sum_mask) for a 64x32 pair tile.
__global__ void __launch_bounds__(256)
pair_tile_kernel(const int* __restrict__ ids,
                 const float* __restrict__ emb,
                 const float* __restrict__ metric,
                 float* __restrict__ ws) {
    __shared__ __align__(16) float Ei[BLK_I][ROWPAD];
    __shared__ __align__(16) float Ej[BLK_J][ROWPAD];
    __shared__ float wC[16];
    __shared__ float wM[16];

    const int t      = threadIdx.x;
    const int lane   = t & 31;
    const int w      = t >> 5;          // wave 0..7
    const int wi     = w >> 1;          // 0..3 (i sub-tile)
    const int wj     = w & 1;           // 0..1 (j sub-tile)
    const int n      = lane & 15;       // column within 16x16 tile
    const int half   = lane >> 4;       // 0 / 1
    const int base_i = blockIdx.y * BLK_I;
    const int base_j = blockIdx.x * BLK_J;

    // --- stage E tiles into LDS with async-to-LDS DMA (ASYNCcnt) ---
    #pragma unroll
    for (int k = 0; k < 8; ++k) {               // Ei: 64 rows * 32 float4 = 2048 chunks
        int chunk = t + k * 256;
        int r = chunk >> 5, c4 = chunk & 31;
        unsigned voff = (unsigned)(((base_i + r) * D_DIM + c4 * 4) * sizeof(float));
        async_copy16(&Ei[r][c4 * 4], voff, emb);
    }
    #pragma unroll
    for (int k = 0; k < 4; ++k) {               // Ej: 32 rows * 32 float4 = 1024 chunks
        int chunk = t + k * 256;
        int r = chunk >> 5, c4 = chunk & 31;
        unsigned voff = (unsigned)(((base_j + r) * D_DIM + c4 * 4) * sizeof(float));
        async_copy16(&Ej[r][c4 * 4], voff, emb);
    }

    // --- while DMA runs: fetch ids, prefetch the 8 random metric gathers per lane ---
    const int jr  = wj * 16 + n;
    const int idJ = ids[base_j + jr];
    int idI[8];
    #pragma unroll
    for (int v = 0; v < 8; ++v) {
        int m = wi * 16 + v + 8 * half;
        idI[v] = ids[base_i + m];
        __builtin_prefetch(&metric[(size_t)idI[v] * N_LANGS + idJ], 0, 0); // global_prefetch_b8
    }

    wait_async0();
    __syncthreads();

    // --- pairwise mean-L1 distance: lane owns 8 (m,n) pairs in WMMA C layout ---
    float acc[8] = {0.f, 0.f, 0.f, 0.f, 0.f, 0.f, 0.f, 0.f};
    for (int d4 = 0; d4 < D_DIM / 4; ++d4) {
        const float4 ej = *(const float4*)&Ej[jr][d4 * 4];
        #pragma unroll
        for (int v = 0; v < 8; ++v) {
            const int ir = wi * 16 + v + 8 * half;
            const float4 ei = *(const float4*)&Ei[ir][d4 * 4];
            acc[v] += fabsf(ei.x - ej.x) + fabsf(ei.y - ej.y) +
                      fabsf(ei.z - ej.z) + fabsf(ei.w - ej.w);
        }
    }

    // --- gather metric, masked |ed - md| contribution ---
    float sumC = 0.f, sumM = 0.f;
    #pragma unroll
    for (int v = 0; v < 8; ++v) {
        float ed = acc[v] * (1.0f / (float)D_DIM);
        float md = metric[(size_t)idI[v] * N_LANGS + idJ];
        float mk = (idI[v] != idJ) ? 1.0f : 0.0f;
        sumC += fabsf(ed - md) * mk;
        sumM += mk;
    }

    // --- wave reduction via WMMA f32 16x16x4 ---
    // A[m][0]=sumC_m, A[m][1]=sumM_m (lanes 0-15); A[m][2]=sumC_{m+16}, A[m][3]=sumM_{m+16}.
    // B selects: col0 sums K=0,2 (contrib), col1 sums K=1,3 (mask).
    v2f A; A.x = sumC; A.y = sumM;
    v2f B; B.x = (n == 0) ? 1.0f : 0.0f; B.y = (n == 1) ? 1.0f : 0.0f;
    v8f C = {};
    v8f Dv = __builtin_amdgcn_wmma_f32_16x16x4_f32(false, A, false, B,
                                                   (short)0, C, false, false);
    float tsum = Dv[0] + Dv[1] + Dv[2] + Dv[3] + Dv[4] + Dv[5] + Dv[6] + Dv[7];
    // lane 0/16 hold contrib halves (col 0), lane 1/17 hold mask halves (col 1)
    if (lane == 0)  wC[w * 2 + 0] = tsum;
    if (lane == 16) wC[w * 2 + 1] = tsum;
    if (lane == 1)  wM[w * 2 + 0] = tsum;
    if (lane == 17) wM[w * 2 + 1] = tsum;
    __syncthreads();

    if (t == 0) {  // fixed-order block combine -> deterministic
        float bc = 0.f, bm = 0.f;
        #pragma unroll
        for (int k = 0; k < 16; ++k) { bc += wC[k]; bm += wM[k]; }
        const int blk = blockIdx.y * gridDim.x + blockIdx.x; // 0..511
        ws[2 * blk + 0] = bc;
        ws[2 * blk + 1] = bm;
    }
}

// Kernel 2: deterministic tree reduction of 512 block partials, final division.
__global__ void __launch_bounds__(256)
final_reduce_kernel(const float* __restrict__ ws, float* __restrict__ out) {
    __shared__ float sc[256];
    __shared__ float sm[256];
    const int t = threadIdx.x;
    sc[t] = ws[2 * t + 0] + ws[2 * (t + 256) + 0];
    sm[t] = ws[2 * t + 1] + ws[2 * (t + 256) + 1];
    __syncthreads();
    for (int s = 128; s > 0; s >>= 1) {
        if (t < s) { sc[t] += sc[t + s]; sm[t] += sm[t + s]; }
        __syncthreads();
    }
    if (t == 0) out[0] = sc[0] / sm[0];
}

extern "C" void kernel_launch(void* const* d_in, const int* in_sizes, int n_in,
                              void* d_out, int out_size, void* d_ws, size_t ws_size,
                              hipStream_t stream) {
    const int*   ids    = (const int*)d_in[0];
    const float* emb    = (const float*)d_in[1];
    const float* metric = (const float*)d_in[2];
    float*       out    = (float*)d_out;
    float*       ws     = (float*)d_ws;

    dim3 grid(1024 / BLK_J, 1024 / BLK_I, 1);  // (32, 16)
    pair_tile_kernel<<<grid, 256, 0, stream>>>(ids, emb, metric, ws);
    final_reduce_kernel<<<1, 256, 0, stream>>>(ws, out);
}